// DifferentiableBiquadChain_3908420239558
// MI455X (gfx1250) — compile-verified
//
#include <hip/hip_runtime.h>

// ---------------------------------------------------------------------------
// Differentiable biquad chain (16 ch x 262144 samples x 16 cascaded biquads)
// Blocked linear-recurrence scan + fp32 WMMA triangular-Toeplitz matmul.
// One workgroup (1024 thr = 32 wave32) per channel; in-place on d_out.
// Phase C: Y(64x16) = T(64x64)*X(64x16) + [c1|c2]*V(2x16), all on the matrix
// pipe (44x v_wmma_f32_16x16x4_f32 per panel, Toeplitz-deduped A operands).
// ---------------------------------------------------------------------------

typedef float v2f __attribute__((ext_vector_type(2)));
typedef float v8f __attribute__((ext_vector_type(8)));

#define NCH    16
#define NSAMP  262144
#define NB     16
#define LCH    64                 // chunk length
#define NCHUNK (NSAMP / LCH)      // 4096
#define TPB    1024
#define CPT    (NCHUNK / TPB)     // 4 chunks per thread
#define NPANEL (NCHUNK / 16)      // 256 panels of 16 chunks
#define NWAVE  (TPB / 32)         // 32 waves

__device__ __forceinline__ float dlog_denorm(float n, float lo, float hi) {
  return __expf(__logf(lo) + n * (__logf(hi) - __logf(lo)));
}

__global__ __launch_bounds__(TPB, 1)
void eq_chain_kernel(const float* __restrict__ audio,
                     const float* __restrict__ params,
                     float* __restrict__ out) {
  const int ch = blockIdx.x;
  const int t  = threadIdx.x;

  __shared__ float coefS[NB][5];     // B0 B1 B2 A1 A2 (a0-normalized)
  __shared__ float MgS[NB][6];       // m00 m01 m10 m11 g1 g2
  __shared__ float hS[LCH];          // impulse response of current stage
  __shared__ float c1S[LCH], c2S[LCH];
  __shared__ float k1S[LCH], k2S[LCH];
  __shared__ float MpowS[LCH][4];    // M^t
  __shared__ float PS[4];            // M^L
  __shared__ float gS[2];            // in/out linear gains
  __shared__ float2 scanS[TPB];
  __shared__ float2 vinS[NCHUNK];    // per-chunk incoming state

  const float* p = params + (size_t)ch * 50;

  // ---- per-stage biquad coefficients ----
  if (t < NB) {
    const float fn = p[t * 3 + 0], gn = p[t * 3 + 1], qn = p[t * 3 + 2];
    const float Q    = dlog_denorm(qn, 0.5f, 16.0f);
    const float gain = -24.0f + gn * 48.0f;
    const float A    = __expf(gain * 0.057564627324851f);   // 10^(g/40)
    const float K2PI = 6.28318530717958647692f / 96000.0f;
    float b0, b1, b2, a0, a1, a2;
    if (t == 0) {                          // highpass
      float w0 = K2PI * dlog_denorm(fn, 20.0f, 500.0f);
      float sn = __sinf(w0), cs = __cosf(w0), al = sn / (2.0f * Q);
      b0 = (1.0f + cs) * 0.5f; b1 = -(1.0f + cs); b2 = b0;
      a0 = 1.0f + al; a1 = -2.0f * cs; a2 = 1.0f - al;
    } else if (t == 15) {                  // lowpass
      float w0 = K2PI * dlog_denorm(fn, 5000.0f, 20000.0f);
      float sn = __sinf(w0), cs = __cosf(w0), al = sn / (2.0f * Q);
      b0 = (1.0f - cs) * 0.5f; b1 = 1.0f - cs; b2 = b0;
      a0 = 1.0f + al; a1 = -2.0f * cs; a2 = 1.0f - al;
    } else if (t == 1) {                   // lowshelf
      float w0 = K2PI * dlog_denorm(fn, 50.0f, 16000.0f);
      float sn = __sinf(w0), cs = __cosf(w0), al = sn / (2.0f * Q);
      float sA = sqrtf(A), tw = 2.0f * sA * al;
      b0 = A * ((A + 1.0f) - (A - 1.0f) * cs + tw);
      b1 = 2.0f * A * ((A - 1.0f) - (A + 1.0f) * cs);
      b2 = A * ((A + 1.0f) - (A - 1.0f) * cs - tw);
      a0 = (A + 1.0f) + (A - 1.0f) * cs + tw;
      a1 = -2.0f * ((A - 1.0f) + (A + 1.0f) * cs);
      a2 = (A + 1.0f) + (A - 1.0f) * cs - tw;
    } else if (t == 14) {                  // highshelf
      float w0 = K2PI * dlog_denorm(fn, 50.0f, 16000.0f);
      float sn = __sinf(w0), cs = __cosf(w0), al = sn / (2.0f * Q);
      float sA = sqrtf(A), tw = 2.0f * sA * al;
      b0 = A * ((A + 1.0f) + (A - 1.0f) * cs + tw);
      b1 = -2.0f * A * ((A - 1.0f) + (A + 1.0f) * cs);
      b2 = A * ((A + 1.0f) + (A - 1.0f) * cs - tw);
      a0 = (A + 1.0f) - (A - 1.0f) * cs + tw;
      a1 = 2.0f * ((A - 1.0f) - (A + 1.0f) * cs);
      a2 = (A + 1.0f) - (A - 1.0f) * cs - tw;
    } else {                               // peak
      float w0 = K2PI * dlog_denorm(fn, 100.0f, 15000.0f);
      float sn = __sinf(w0), cs = __cosf(w0), al = sn / (2.0f * Q);
      b0 = 1.0f + al * A; b1 = -2.0f * cs; b2 = 1.0f - al * A;
      a0 = 1.0f + al / A; a1 = b1;         a2 = 1.0f - al / A;
    }
    const float inv = 1.0f / a0;
    const float B0 = b0 * inv, B1 = b1 * inv, B2 = b2 * inv;
    const float A1 = a1 * inv, A2 = a2 * inv;
    coefS[t][0] = B0; coefS[t][1] = B1; coefS[t][2] = B2;
    coefS[t][3] = A1; coefS[t][4] = A2;
    MgS[t][0] = -A1; MgS[t][1] = 1.0f; MgS[t][2] = -A2; MgS[t][3] = 0.0f;
    MgS[t][4] = B1 - A1 * B0;            // g1
    MgS[t][5] = B2 - A2 * B0;            // g2
  }
  if (t == 16) gS[0] = __expf((-60.0f + p[48] * 60.0f) * 0.11512925464970f);
  if (t == 17) gS[1] = __expf((-60.0f + p[49] * 60.0f) * 0.11512925464970f);
  __syncthreads();

  // ---- pre-pass: out = audio * in_gain (vectorized) ----
  float* buf = out + (size_t)ch * NSAMP;
  {
    const float gin = gS[0];
    const float4* s4 = (const float4*)(audio + (size_t)ch * NSAMP);
    float4* d4 = (float4*)buf;
    for (int i = t; i < NSAMP / 4; i += TPB) {
      float4 v = s4[i];
      v.x *= gin; v.y *= gin; v.z *= gin; v.w *= gin;
      d4[i] = v;
    }
  }
  __syncthreads();

  // ---- cascade of 16 stages ----
  for (int s = 0; s < NB; ++s) {
    // --- stage setup: impulse response h[t], M^t powers, P = M^L ---
    if (t == 0) {
      const float B0 = coefS[s][0], B1 = coefS[s][1], B2 = coefS[s][2];
      const float A1 = coefS[s][3], A2 = coefS[s][4];
      float s1 = 0.0f, s2 = 0.0f;
      for (int j = 0; j < LCH; ++j) {
        const float x = (j == 0) ? 1.0f : 0.0f;
        const float y = fmaf(B0, x, s1);
        hS[j] = y;
        const float ns1 = fmaf(B1, x, fmaf(-A1, y, s2));
        s2 = fmaf(B2, x, -A2 * y);
        s1 = ns1;
      }
    } else if (t == 1) {
      const float m00 = MgS[s][0], m01 = MgS[s][1];
      const float m10 = MgS[s][2], m11 = MgS[s][3];
      float c00 = 1.0f, c01 = 0.0f, c10 = 0.0f, c11 = 1.0f;
      for (int j = 0; j < LCH; ++j) {
        MpowS[j][0] = c00; MpowS[j][1] = c01; MpowS[j][2] = c10; MpowS[j][3] = c11;
        const float n00 = m00 * c00 + m01 * c10, n01 = m00 * c01 + m01 * c11;
        const float n10 = m10 * c00 + m11 * c10, n11 = m10 * c01 + m11 * c11;
        c00 = n00; c01 = n01; c10 = n10; c11 = n11;
      }
      PS[0] = c00; PS[1] = c01; PS[2] = c10; PS[3] = c11;
    }
    __syncthreads();
    if (t < LCH) {
      c1S[t] = MpowS[t][0];
      c2S[t] = MpowS[t][1];
      const float g1 = MgS[s][4], g2 = MgS[s][5];
      const int j = LCH - 1 - t;
      k1S[t] = MpowS[j][0] * g1 + MpowS[j][1] * g2;
      k2S[t] = MpowS[j][2] * g1 + MpowS[j][3] * g2;
    }
    __syncthreads();

    // --- Phase A: per-chunk state aggregates w[c] = K * x_chunk ---
    float2 w[CPT];
    for (int i = 0; i < CPT; ++i) {
      const float* xc = buf + (size_t)(t * CPT + i) * LCH;
      float w1 = 0.0f, w2 = 0.0f;
      #pragma unroll 8
      for (int j = 0; j < LCH; ++j) {
        const float xv = xc[j];
        w1 = fmaf(k1S[j], xv, w1);
        w2 = fmaf(k2S[j], xv, w2);
      }
      w[i] = make_float2(w1, w2);
    }

    const float p00 = PS[0], p01 = PS[1], p10 = PS[2], p11 = PS[3];
    float2 agg = make_float2(0.0f, 0.0f);
    #pragma unroll
    for (int i = 0; i < CPT; ++i) {
      const float ax = p00 * agg.x + p01 * agg.y + w[i].x;
      const float ay = p10 * agg.x + p11 * agg.y + w[i].y;
      agg = make_float2(ax, ay);
    }

    // --- Phase B: Hillis-Steele scan over thread groups (weight P^4) ---
    float q00, q01, q10, q11;
    {
      const float r00 = p00 * p00 + p01 * p10, r01 = p00 * p01 + p01 * p11;
      const float r10 = p10 * p00 + p11 * p10, r11 = p10 * p01 + p11 * p11;
      q00 = r00 * r00 + r01 * r10; q01 = r00 * r01 + r01 * r11;
      q10 = r10 * r00 + r11 * r10; q11 = r10 * r01 + r11 * r11;
    }
    float2 cur = agg;
    for (int d = 0; d < 10; ++d) {
      scanS[t] = cur;
      __syncthreads();
      const int off = 1 << d;
      if (t >= off) {
        const float2 o = scanS[t - off];
        cur.x += q00 * o.x + q01 * o.y;
        cur.y += q10 * o.x + q11 * o.y;
      }
      __syncthreads();
      const float n00 = q00 * q00 + q01 * q10, n01 = q00 * q01 + q01 * q11;
      const float n10 = q10 * q00 + q11 * q10, n11 = q10 * q01 + q11 * q11;
      q00 = n00; q01 = n01; q10 = n10; q11 = n11;
    }
    scanS[t] = cur;
    __syncthreads();
    float2 vv = (t > 0) ? scanS[t - 1] : make_float2(0.0f, 0.0f);
    #pragma unroll
    for (int i = 0; i < CPT; ++i) {
      vinS[t * CPT + i] = vv;
      const float nx = p00 * vv.x + p01 * vv.y + w[i].x;
      const float ny = p10 * vv.x + p11 * vv.y + w[i].y;
      vv = make_float2(nx, ny);
    }
    __syncthreads();

    // --- Phase C: Y = T*X + [c1|c2]*V entirely on the matrix pipe ---
    const int wave = t >> 5, lane = t & 31;
    const int m = lane & 15, hi = lane >> 4;   // half-wave K split
    const float oscale = (s == NB - 1) ? gS[1] : 1.0f;

    // A operands are Toeplitz at tile level too: tile(r,kk) == diag d=r-kk.
    // Only 16 distinct pairs + 4 state-response pairs stay live across panels.
    v2f Ad[4][4];
    #pragma unroll
    for (int d = 0; d < 4; ++d) {
      #pragma unroll
      for (int q = 0; q < 4; ++q) {
        const int i0 = d * 16 + m - (4 * q + 2 * hi);   // always <= 63
        Ad[d][q].x = (i0 >= 0) ? hS[i0] : 0.0f;
        Ad[d][q].y = (i0 >= 1) ? hS[i0 - 1] : 0.0f;
      }
    }
    v2f Ac[4];                                 // [c1 | c2 | 0 | 0] columns
    #pragma unroll
    for (int r = 0; r < 4; ++r) {
      Ac[r].x = hi ? 0.0f : c1S[r * 16 + m];
      Ac[r].y = hi ? 0.0f : c2S[r * 16 + m];
    }

    for (int pp = wave; pp < NPANEL; pp += NWAVE) {
      float* pb = buf + (size_t)pp * (16 * LCH);
      if (pp + NWAVE < NPANEL)                 // global_prefetch_b8 next panel
        __builtin_prefetch(buf + (size_t)(pp + NWAVE) * (16 * LCH) + m * LCH, 0, 1);

      const v8f vzero = {0.f, 0.f, 0.f, 0.f, 0.f, 0.f, 0.f, 0.f};
      v8f acc[4] = {vzero, vzero, vzero, vzero};
      #pragma unroll
      for (int kk = 0; kk < 4; ++kk) {
        v2f Bv[4];
        #pragma unroll
        for (int q = 0; q < 4; ++q) {
          const float2 b2v = *(const float2*)(pb + m * LCH + kk * 16 + 4 * q + 2 * hi);
          Bv[q].x = b2v.x; Bv[q].y = b2v.y;
        }
        #pragma unroll
        for (int r = 0; r < 4; ++r) {
          if (r < kk) continue;                // upper tiles of T are zero
          #pragma unroll
          for (int q = 0; q < 4; ++q) {
            acc[r] = __builtin_amdgcn_wmma_f32_16x16x4_f32(
                false, Ad[r - kk][q], false, Bv[q], (short)0, acc[r], false, false);
          }
        }
      }
      // state response as one more rank-4 update per row tile
      {
        const float2 vi = vinS[pp * 16 + m];
        v2f Bx;
        Bx.x = hi ? 0.0f : vi.x;
        Bx.y = hi ? 0.0f : vi.y;
        #pragma unroll
        for (int r = 0; r < 4; ++r) {
          acc[r] = __builtin_amdgcn_wmma_f32_16x16x4_f32(
              false, Ac[r], false, Bx, (short)0, acc[r], false, false);
        }
      }
      // epilogue: scale (final stage folds out_gain) + vector store, in-place
      #pragma unroll
      for (int r = 0; r < 4; ++r) {
        float4 f0, f1;
        f0.x = acc[r][0] * oscale; f0.y = acc[r][1] * oscale;
        f0.z = acc[r][2] * oscale; f0.w = acc[r][3] * oscale;
        f1.x = acc[r][4] * oscale; f1.y = acc[r][5] * oscale;
        f1.z = acc[r][6] * oscale; f1.w = acc[r][7] * oscale;
        float* dst = pb + m * LCH + r * 16 + 8 * hi;
        *(float4*)dst = f0;
        *(float4*)(dst + 4) = f1;
      }
    }
    __syncthreads();
  }
}

extern "C" void kernel_launch(void* const* d_in, const int* in_sizes, int n_in,
                              void* d_out, int out_size, void* d_ws, size_t ws_size,
                              hipStream_t stream) {
  (void)in_sizes; (void)n_in; (void)d_ws; (void)ws_size; (void)out_size;
  const float* audio  = (const float*)d_in[0];
  const float* params = (const float*)d_in[1];
  float* out = (float*)d_out;
  eq_chain_kernel<<<dim3(NCH), dim3(TPB), 0, stream>>>(audio, params, out);
}